// GCN_68118181314618
// MI455X (gfx1250) — compile-verified
//
#include <hip/hip_runtime.h>
#include <hip/hip_bf16.h>

// ---------------------------------------------------------------------------
// GCN (3-layer) for MI455X / gfx1250.
//   h   = X @ W                      (fp32 WMMA 16x16x4, A via LDS, B via Wt)
//   agg = segment_sum(h[src]*norm)   (wave-per-edge fp32 global atomics)
//   out = relu(agg + h*deg_inv + b)
// ---------------------------------------------------------------------------

typedef __attribute__((ext_vector_type(2))) float v2f;
typedef __attribute__((ext_vector_type(8))) float v8f;

// ---------------------------------------------------------------------------
// Transpose W[K x 128] -> Wt[128 x K] so the two K-adjacent B elements each
// WMMA needs are contiguous (single global_load_b64 per step per lane).
// ---------------------------------------------------------------------------
__global__ void gcn_transpose_w(const float* __restrict__ W,
                                float* __restrict__ Wt, int K) {
  int i = blockIdx.x * blockDim.x + threadIdx.x;
  if (i < K * 128) {
    const int k = i >> 7;
    const int c = i & 127;
    Wt[(size_t)c * K + k] = W[i];
  }
}

// ---------------------------------------------------------------------------
// WMMA GEMM: H[nrows x 128] = X[nrows x K] @ W[K x 128], W given transposed
// as Wt[128 x K]. One block = 256 threads = 8 waves. Block computes a 16-row
// stripe; wave w computes the 16x16 tile at columns [16w, 16w+16).
// ---------------------------------------------------------------------------
template <int K>
__global__ __launch_bounds__(256) void gcn_gemm_wmma(
    const float* __restrict__ X, const float* __restrict__ Wt,
    float* __restrict__ H, int nrows) {
  constexpr int LDA = K + 4;  // pad to avoid 16-way LDS bank conflicts
  __shared__ __align__(16) float As[16 * LDA];

  const int rowBase = blockIdx.x * 16;
  const int tid = threadIdx.x;

  // Cooperative load of the 16 x K A-tile into LDS (float4 granularity).
  constexpr int vecPerRow = K / 4;
  for (int i = tid; i < 16 * vecPerRow; i += 256) {
    const int r = i / vecPerRow;
    const int v = i % vecPerRow;
    const int grow = rowBase + r;
    float4 val = make_float4(0.f, 0.f, 0.f, 0.f);
    if (grow < nrows) {
      val = *reinterpret_cast<const float4*>(X + (size_t)grow * K + 4 * v);
    }
    *reinterpret_cast<float4*>(&As[r * LDA + 4 * v]) = val;
  }
  __syncthreads();

  const int wave = tid >> 5;          // 0..7  -> column tile
  const int lane = tid & 31;
  const int l15 = lane & 15;
  const int halfSel = lane >> 4;      // 0: K pair {k,k+1}; 1: {k+2,k+3}
  const int col = wave * 16 + l15;

  const float* __restrict__ Bcol = Wt + (size_t)col * K;  // contiguous in K

  v8f acc = {};
#pragma unroll 4
  for (int k0 = 0; k0 < K; k0 += 4) {
    const int kk = k0 + 2 * halfSel;
    // A 16x4: lanes 0-15 hold row=l15, K={k0,k0+1}; lanes 16-31 K={k0+2,k0+3}
    v2f a;
    a.x = As[l15 * LDA + kk];
    a.y = As[l15 * LDA + kk + 1];
    // B 4x16: lanes 0-15 hold col, K={k0,k0+1}; lanes 16-31 K={k0+2,k0+3}
    v2f b;
    b.x = Bcol[kk];
    b.y = Bcol[kk + 1];
    acc = __builtin_amdgcn_wmma_f32_16x16x4_f32(
        /*neg_a=*/false, a, /*neg_b=*/false, b,
        /*c_mod=*/(short)0, acc, /*reuse_a=*/false, /*reuse_b=*/false);
  }

  // C/D layout: VGPR r holds M = r + 8*halfSel, N = col.
  if (rowBase + 16 <= nrows) {  // uniform fast path: full tile, no exec churn
    float* __restrict__ Hp = H + (size_t)(rowBase + 8 * halfSel) * 128 + col;
#pragma unroll
    for (int r = 0; r < 8; ++r) {
      Hp[(size_t)r * 128] = acc[r];
    }
  } else {  // tail block only
#pragma unroll
    for (int r = 0; r < 8; ++r) {
      const int m = rowBase + r + 8 * halfSel;
      if (m < nrows) H[(size_t)m * 128 + col] = acc[r];
    }
  }
}

// ---------------------------------------------------------------------------
// Degree kernels
// ---------------------------------------------------------------------------
__global__ void gcn_deg_init(float* __restrict__ deg, int n) {
  int i = blockIdx.x * blockDim.x + threadIdx.x;
  if (i < n) deg[i] = 1.0f;  // self loop
}

__global__ void gcn_deg_count(const int* __restrict__ dst,
                              float* __restrict__ deg, int e) {
  int i = blockIdx.x * blockDim.x + threadIdx.x;
  if (i < e) atomicAdd(&deg[dst[i]], 1.0f);
}

__global__ void gcn_deg_finalize(const float* __restrict__ deg,
                                 float* __restrict__ dis,
                                 float* __restrict__ dinv, int n) {
  int i = blockIdx.x * blockDim.x + threadIdx.x;
  if (i < n) {
    float d = deg[i];
    dis[i] = rsqrtf(d);
    dinv[i] = 1.0f / d;
  }
}

__global__ void gcn_zero(float* __restrict__ p, int n) {
  int i = blockIdx.x * blockDim.x + threadIdx.x;
  if (i < n) p[i] = 0.0f;
}

// ---------------------------------------------------------------------------
// Edge scatter: one wave (32 lanes) per edge; each lane moves 4 features.
// agg[dst] += h[src] * (deg_inv_sqrt[src] * deg_inv_sqrt[dst])
// ---------------------------------------------------------------------------
__global__ __launch_bounds__(256) void gcn_scatter(
    const int* __restrict__ src, const int* __restrict__ dst,
    const float* __restrict__ H, const float* __restrict__ dis,
    float* __restrict__ agg, int nedges) {
  const int e = blockIdx.x * 8 + (threadIdx.x >> 5);
  if (e >= nedges) return;
  const int lane = threadIdx.x & 31;
  const int s = src[e];
  const int d = dst[e];
  const float norm = dis[s] * dis[d];
  const float4 v =
      reinterpret_cast<const float4*>(H + (size_t)s * 128)[lane];
  float* ad = agg + (size_t)d * 128 + lane * 4;
  atomicAdd(ad + 0, v.x * norm);
  atomicAdd(ad + 1, v.y * norm);
  atomicAdd(ad + 2, v.z * norm);
  atomicAdd(ad + 3, v.w * norm);
}

// ---------------------------------------------------------------------------
// Combine: out = relu(agg + h*deg_inv + b)
// ---------------------------------------------------------------------------
__global__ void gcn_combine(const float* __restrict__ agg,
                            const float* __restrict__ H,
                            const float* __restrict__ dinv,
                            const float* __restrict__ bias,
                            float* __restrict__ out, int total) {
  int i = blockIdx.x * blockDim.x + threadIdx.x;
  if (i < total) {
    const int node = i >> 7;
    const int f = i & 127;
    const float v = agg[i] + H[i] * dinv[node] + bias[f];
    out[i] = fmaxf(v, 0.0f);
  }
}

// ---------------------------------------------------------------------------
extern "C" void kernel_launch(void* const* d_in, const int* in_sizes, int n_in,
                              void* d_out, int out_size, void* d_ws,
                              size_t ws_size, hipStream_t stream) {
  const float* x = (const float*)d_in[0];
  const int* edge = (const int*)d_in[1];
  const float* W0 = (const float*)d_in[2];
  const float* b0 = (const float*)d_in[3];
  const float* W1 = (const float*)d_in[4];
  const float* b1 = (const float*)d_in[5];
  const float* W2 = (const float*)d_in[6];
  const float* b2 = (const float*)d_in[7];
  float* out = (float*)d_out;

  const int N = in_sizes[0] / 256;   // 50000
  const int E = in_sizes[1] / 2;     // 800000
  const int* srcIdx = edge;
  const int* dstIdx = edge + E;

  float* ws = (float*)d_ws;
  float* hbuf = ws;                          // N*128
  float* aggbuf = hbuf + (size_t)N * 128;    // N*128
  float* xin = aggbuf + (size_t)N * 128;     // N*128 (layer activations)
  float* deg = xin + (size_t)N * 128;        // N
  float* dis = deg + N;                      // N
  float* dinv = dis + N;                     // N
  float* wt0 = dinv + N;                     // 128*256 (W0^T)
  float* wt1 = wt0 + 128 * 256;              // 128*128 (W1^T)
  float* wt2 = wt1 + 128 * 128;              // 128*128 (W2^T)

  const int T = 256;
  const int nodeBlocks = (N + T - 1) / T;
  const int featTotal = N * 128;
  const int featBlocks = (featTotal + T - 1) / T;
  const int edgeBlocks = (E + T - 1) / T;
  const int scatBlocks = (E + 7) / 8;
  const int gemmBlocks = (N + 15) / 16;

  // weight transposes (tiny, once per launch)
  gcn_transpose_w<<<(256 * 128 + T - 1) / T, T, 0, stream>>>(W0, wt0, 256);
  gcn_transpose_w<<<(128 * 128 + T - 1) / T, T, 0, stream>>>(W1, wt1, 128);
  gcn_transpose_w<<<(128 * 128 + T - 1) / T, T, 0, stream>>>(W2, wt2, 128);

  // degrees (with self loops), normalizers
  gcn_deg_init<<<nodeBlocks, T, 0, stream>>>(deg, N);
  gcn_deg_count<<<edgeBlocks, T, 0, stream>>>(dstIdx, deg, E);
  gcn_deg_finalize<<<nodeBlocks, T, 0, stream>>>(deg, dis, dinv, N);

  // ---- layer 0: 256 -> 128 ----
  gcn_gemm_wmma<256><<<gemmBlocks, T, 0, stream>>>(x, wt0, hbuf, N);
  gcn_zero<<<featBlocks, T, 0, stream>>>(aggbuf, featTotal);
  gcn_scatter<<<scatBlocks, T, 0, stream>>>(srcIdx, dstIdx, hbuf, dis, aggbuf, E);
  gcn_combine<<<featBlocks, T, 0, stream>>>(aggbuf, hbuf, dinv, b0, xin, featTotal);

  // ---- layer 1: 128 -> 128 ----
  gcn_gemm_wmma<128><<<gemmBlocks, T, 0, stream>>>(xin, wt1, hbuf, N);
  gcn_zero<<<featBlocks, T, 0, stream>>>(aggbuf, featTotal);
  gcn_scatter<<<scatBlocks, T, 0, stream>>>(srcIdx, dstIdx, hbuf, dis, aggbuf, E);
  gcn_combine<<<featBlocks, T, 0, stream>>>(aggbuf, hbuf, dinv, b1, xin, featTotal);

  // ---- layer 2: 128 -> 128 ----
  gcn_gemm_wmma<128><<<gemmBlocks, T, 0, stream>>>(xin, wt2, hbuf, N);
  gcn_zero<<<featBlocks, T, 0, stream>>>(aggbuf, featTotal);
  gcn_scatter<<<scatBlocks, T, 0, stream>>>(srcIdx, dstIdx, hbuf, dis, aggbuf, E);
  gcn_combine<<<featBlocks, T, 0, stream>>>(aggbuf, hbuf, dinv, b2, out, featTotal);
}